// RoiPoolingConv_36661840839578
// MI455X (gfx1250) — compile-verified
//
#include <hip/hip_runtime.h>
#include <cstdint>

// ROI bilinear crop-resize (TF2 crop_and_resize semantics), MI455X / gfx1250.
// img : (1, 200, 200, 1024) f32, rois : (1, 300, 4) f32 [x,y,w,h]
// out : (1, 300, 7, 7, 1024) f32
//
// Memory-bound kernel: 60 MB streamed out (non-temporal), gathers stay in the
// 192 MB L2 (image = 164 MB). Gathers use the CDNA5 async global->LDS path
// (ASYNCcnt) so the wave issues all 4 gathers without VGPR return dependencies.

#define POOL 7
#define IMG_W 200
#define IMG_C 1024
#define NROI 300

typedef __attribute__((ext_vector_type(4))) float v4f;

__device__ __forceinline__ void async_gather_b128(uint32_t lds_off, const float* gaddr) {
    // VDST = per-lane LDS byte address, VADDR = per-lane 64-bit global address (GV mode)
    asm volatile("global_load_async_to_lds_b128 %0, %1, off"
                 :
                 : "v"(lds_off), "v"(gaddr)
                 : "memory");
}

__device__ __forceinline__ void wait_async0() {
#if __has_builtin(__builtin_amdgcn_s_wait_asynccnt)
    __builtin_amdgcn_s_wait_asynccnt(0);
#else
    asm volatile("s_wait_asynccnt 0" ::: "memory");
#endif
}

// start/size -> clipped low/high gather coords + lerp frac (TF2 half-pixel)
__device__ __forceinline__ void axis_coords(int j, int start, int size,
                                            int& i0c, int& i1c, float& frac) {
    float s = ((float)j + 0.5f) * (float)size / 7.0f - 0.5f;
    int i0 = (int)floorf(s);
    frac = s - (float)i0;
    int hi = size - 1;
    i0c = min(max(i0, 0), hi) + start;
    i1c = min(max(i0 + 1, 0), hi) + start;
}

__global__ __launch_bounds__(256) void roi_bilinear_kernel(
    const float* __restrict__ img,   // (200*200, 1024)
    const float* __restrict__ rois,  // (300, 4)
    float* __restrict__ out)         // (300*49, 1024)
{
    __shared__ v4f smem[4][256];     // v00 / v01 / v10 / v11 staging, 16 KB

    const int t   = threadIdx.x;
    const int bid = blockIdx.x;                 // 0 .. 300*49-1
    const int r   = bid / (POOL * POOL);
    const int p   = bid - r * (POOL * POOL);
    const int py  = p / POOL;
    const int px  = p - py * POOL;

    // ROI params (uniform per block -> scalarized by compiler)
    const float rx = rois[r * 4 + 0];
    const float ry = rois[r * 4 + 1];
    const float rw = rois[r * 4 + 2];
    const float rh = rois[r * 4 + 3];
    const int xi = (int)rx, yi = (int)ry, wi = (int)rw, hi = (int)rh;

    int y0, y1, x0, x1;
    float fy, fx;
    axis_coords(py, yi, hi, y0, y1, fy);
    axis_coords(px, xi, wi, x0, x1, fx);

    const int c0 = t * 4;                       // 4 channels per thread
    const size_t row00 = (size_t)(y0 * IMG_W + x0) * IMG_C + c0;
    const size_t row01 = (size_t)(y0 * IMG_W + x1) * IMG_C + c0;
    const size_t row10 = (size_t)(y1 * IMG_W + x0) * IMG_C + c0;
    const size_t row11 = (size_t)(y1 * IMG_W + x1) * IMG_C + c0;

    // Issue all 4 gathers as async global->LDS b128 ops (one ASYNCcnt each per wave)
    async_gather_b128((uint32_t)(uintptr_t)&smem[0][t], img + row00);
    async_gather_b128((uint32_t)(uintptr_t)&smem[1][t], img + row01);
    async_gather_b128((uint32_t)(uintptr_t)&smem[2][t], img + row10);
    async_gather_b128((uint32_t)(uintptr_t)&smem[3][t], img + row11);

    wait_async0();  // each thread reads only slots its own wave loaded -> no barrier

    const v4f v00 = smem[0][t];
    const v4f v01 = smem[1][t];
    const v4f v10 = smem[2][t];
    const v4f v11 = smem[3][t];

    const float gx = 1.0f - fx, gy = 1.0f - fy;
    v4f top = v00 * gx + v01 * fx;
    v4f bot = v10 * gx + v11 * fx;
    v4f res = top * gy + bot * fy;

    // Streaming 60 MB output: non-temporal so the L2 keeps the image resident.
    __builtin_nontemporal_store(res, (v4f*)(out + (size_t)bid * IMG_C + c0));
}

extern "C" void kernel_launch(void* const* d_in, const int* in_sizes, int n_in,
                              void* d_out, int out_size, void* d_ws, size_t ws_size,
                              hipStream_t stream) {
    const float* img  = (const float*)d_in[0];   // (1,200,200,1024) f32
    const float* rois = (const float*)d_in[1];   // (1,300,4) f32
    float* out = (float*)d_out;                  // (1,300,7,7,1024) f32

    const int nblocks = NROI * POOL * POOL;      // 14700
    roi_bilinear_kernel<<<nblocks, 256, 0, stream>>>(img, rois, out);
    (void)in_sizes; (void)n_in; (void)out_size; (void)d_ws; (void)ws_size;
}